// CudaMorphUnpool2D_32727650796053
// MI455X (gfx1250) — compile-verified
//
#include <hip/hip_runtime.h>
#include <cstdint>

// Problem constants (match reference)
#define BB   16
#define CC   64
#define HH   256
#define WW   256
#define HPc  128
#define WPc  128

// Tiling
#define TH   64          // output tile height
#define TW   64          // output tile width
#define CH_  66          // canvas rows/cols (tile + 1 halo each side)
#define CW_  68          // canvas row stride (66 used, padded)
#define PR   34          // pooled rows/cols covering tile + halo (always in-bounds)
#define NPE  (PR * PR)   // 1156 pooled elements per tile
#define NTHREADS 256

typedef unsigned int v4u __attribute__((ext_vector_type(4)));
typedef int          v8i __attribute__((ext_vector_type(8)));
typedef int          v4i __attribute__((ext_vector_type(4)));

__global__ __launch_bounds__(NTHREADS)
void morph_unpool_dilate_kernel(const float* __restrict__ f,
                                const int*   __restrict__ prov,
                                float*       __restrict__ out)
{
    __shared__ float canvas[CH_ * CW_];   // 66x68 floats
    __shared__ float FS[NPE];             // staged pooled values (TDM dest)
    __shared__ int   PS[NPE];             // staged provenance    (TDM dest)

    const int tid = threadIdx.x;
    const int blk = blockIdx.x;
    const int bc  = blk >> 4;                 // plane index (1024 planes)
    const int t   = blk & 15;                 // 4x4 tiles per plane
    const int y0  = (t >> 2) << 6;
    const int x0  = (t & 3)  << 6;

    // Pooled-tile origin, clamped so the 34x34 tile is always fully in-bounds.
    int i0 = (y0 >> 1) - 1; if (i0 < 0) i0 = 0; if (i0 > HPc - PR) i0 = HPc - PR;
    int j0 = (x0 >> 1) - 1; if (j0 < 0) j0 = 0; if (j0 > WPc - PR) j0 = WPc - PR;

    const size_t planeP = (size_t)bc * (HPc * WPc);
    const size_t planeO = (size_t)bc * (HH * WW);
    const size_t off0   = planeP + (size_t)i0 * WPc + (size_t)j0;

    // ---- Phase 0: Tensor Data Mover — DMA the 34x34 pooled tile into LDS ----
    // One tensor_load_to_lds per buffer, issued by wave 0 only (TDM ignores
    // EXEC; a whole-wave branch prevents duplicate issue from waves 1..7).
    if (tid < 32) {
        const uint64_t gaF = (uint64_t)(uintptr_t)(f + off0);
        const uint64_t gaP = (uint64_t)(uintptr_t)(prov + off0);
        const uint32_t ldsF = (uint32_t)(uintptr_t)(&FS[0]);
        const uint32_t ldsP = (uint32_t)(uintptr_t)(&PS[0]);

        // D# group1: data_size=2 (4B), dims 128x128, tile 34x34, stride0=128,
        // stride1=16384, no padding / iterate / multicast / atomic-barrier.
        const v8i g1 = { 0x00020000,                 // data_size=4B
                         (int)(128u << 16),          // tensor_dim0 = 128 (lo16)
                         (int)(128u << 16),          // tensor_dim0 hi=0 | tensor_dim1=128 (lo16)
                         (int)(34u  << 16),          // tensor_dim1 hi=0 | tile_dim0 = 34
                         34,                         // tile_dim1 = 34, tile_dim2 = 0
                         128,                        // tensor_dim0_stride = 128
                         (int)(16384u << 16),        // stride0 hi=0 | tensor_dim1_stride lo16
                         0 };                        // tensor_dim1_stride hi
        const v4i z4 = { 0, 0, 0, 0 };
        const v8i z8 = { 0, 0, 0, 0, 0, 0, 0, 0 };

        const v4u g0f = { 1u, ldsF, (uint32_t)gaF,
                          (uint32_t)(gaF >> 32) | 0x80000000u };   // type=2 in [127:126]
        const v4u g0p = { 1u, ldsP, (uint32_t)gaP,
                          (uint32_t)(gaP >> 32) | 0x80000000u };

        __builtin_amdgcn_tensor_load_to_lds(g0f, g1, z4, z4, z8, 0);
        __builtin_amdgcn_tensor_load_to_lds(g0p, g1, z4, z4, z8, 0);
    }

    // ---- Phase 1: init canvas while the TDM transfers are in flight ----
    // In-image cells -> 0 (unpool background); out-of-image halo -> -inf
    // (reduce_window pads with -inf).
    for (int e = tid; e < CH_ * CW_; e += NTHREADS) {
        int lr = e / CW_;
        int lc = e - lr * CW_;
        int gy = y0 - 1 + lr;
        int gx = x0 - 1 + lc;
        bool inimg = (lc < CH_) &&
                     ((unsigned)gy < (unsigned)HH) &&
                     ((unsigned)gx < (unsigned)WW);
        canvas[e] = inimg ? 0.0f : -__builtin_huge_valf();
    }

    if (tid < 32) __builtin_amdgcn_s_wait_tensorcnt(0);
    __syncthreads();

    // ---- Phase 2: scatter pooled values into the LDS canvas ----
    // Window overlaps (stride-2 pooling, 3x3 windows) can collide; resolve
    // deterministically with single-instruction LDS float max atomics.
    const uint32_t canvasBase = (uint32_t)(uintptr_t)(&canvas[0]);
    for (int e = tid; e < NPE; e += NTHREADS) {
        int   p = PS[e];
        float v = FS[e];
        int lr = (p >> 8)  - (y0 - 1);      // W == 256 -> row = p>>8
        int lc = (p & 255) - (x0 - 1);
        if ((unsigned)lr < (unsigned)CH_ && (unsigned)lc < (unsigned)CH_) {
            uint32_t a = canvasBase + (uint32_t)(lr * CW_ + lc) * 4u;
            asm volatile("ds_max_num_f32 %0, %1" :: "v"(a), "v"(v) : "memory");
        }
    }

    __syncthreads();

    // ---- Phase 3: 3x3 windowed max over the canvas, coalesced stores ----
    for (int e = tid; e < TH * TW; e += NTHREADS) {
        int oy = e >> 6;
        int ox = e & 63;
        const float* c0 = &canvas[oy * CW_ + ox];   // canvas[0][0] == (y0-1, x0-1)
        float m;
        m = fmaxf(c0[0],          c0[1]);
        m = fmaxf(m,              c0[2]);
        m = fmaxf(m,              c0[CW_ + 0]);
        m = fmaxf(m,              c0[CW_ + 1]);
        m = fmaxf(m,              c0[CW_ + 2]);
        m = fmaxf(m,              c0[2 * CW_ + 0]);
        m = fmaxf(m,              c0[2 * CW_ + 1]);
        m = fmaxf(m,              c0[2 * CW_ + 2]);
        out[planeO + (size_t)(y0 + oy) * WW + (size_t)(x0 + ox)] = m;
    }
}

extern "C" void kernel_launch(void* const* d_in, const int* in_sizes, int n_in,
                              void* d_out, int out_size, void* d_ws, size_t ws_size,
                              hipStream_t stream)
{
    (void)in_sizes; (void)n_in; (void)d_ws; (void)ws_size; (void)out_size;
    const float* f    = (const float*)d_in[0];
    const int*   prov = (const int*)d_in[1];
    // d_in[2] (h) and d_in[3] (w) are fixed scalars (256, 256) in this problem.
    float* out = (float*)d_out;

    const int nplanes = BB * CC;                 // 1024
    const int tiles   = (HH / TH) * (WW / TW);   // 16
    dim3 grid(nplanes * tiles);                  // 16384 blocks
    dim3 block(NTHREADS);
    morph_unpool_dilate_kernel<<<grid, block, 0, stream>>>(f, prov, out);
}